// KLTTracker_51874615001662
// MI455X (gfx1250) — compile-verified
//
#include <hip/hip_runtime.h>

#define IMH 1080
#define IMW 1920
#define NPTS 8192
#define WIN 25
#define NSAMP (WIN * WIN)
#define LEVELS 15
#define HALFW 12.0f
#define HWPX (IMH * IMW)

typedef float v2f __attribute__((ext_vector_type(2)));
typedef float v8f __attribute__((ext_vector_type(8)));

// Normalized 5-tap Gaussian, sigma=0.2 (exp(-x^2/0.08) / sum)
__constant__ float KW5[5] = {1.9287354e-22f, 3.7266254e-06f, 9.9999255e-01f,
                             3.7266254e-06f, 1.9287354e-22f};

__device__ __forceinline__ int clampi(int v, int lo, int hi) {
    return v < lo ? lo : (v > hi ? hi : v);
}

// ---------------------------------------------------------------------------
// Wave-level sum reduction. One V_WMMA_F32_16X16X4_F32 with A = ones(16x4),
// B-VGPR0 = per-lane partials, B-VGPR1 = 0 gives D[m][n] = p[n] + p[n+16]
// for every m (D VGPR0 lane L holds s[L % 16]); 4 shfl_xor steps finish.
// ---------------------------------------------------------------------------
__device__ __forceinline__ float wave_reduce_sum(float p) {
#if __has_builtin(__builtin_amdgcn_wmma_f32_16x16x4_f32)
    v2f a; a[0] = 1.0f; a[1] = 1.0f;   // A matrix: all ones
    v2f b; b[0] = p;    b[1] = 0.0f;   // B matrix: partials in K rows of VGPR0
    v8f c = {};
    c = __builtin_amdgcn_wmma_f32_16x16x4_f32(
            /*neg_a=*/false, a, /*neg_b=*/false, b,
            /*c_mod=*/(short)0, c, /*reuse_a=*/false, /*reuse_b=*/false);
    float r = c[0];                    // r[L] = p[L%16] + p[(L%16)+16]
    r += __shfl_xor(r, 8, 32);
    r += __shfl_xor(r, 4, 32);
    r += __shfl_xor(r, 2, 32);
    r += __shfl_xor(r, 1, 32);
    return r;                          // total replicated to all lanes
#else
    float r = p;
    r += __shfl_xor(r, 16, 32);
    r += __shfl_xor(r, 8, 32);
    r += __shfl_xor(r, 4, 32);
    r += __shfl_xor(r, 2, 32);
    r += __shfl_xor(r, 1, 32);
    return r;
#endif
}

// Block-wide sum over 256 threads (8 waves); result returned to every thread.
__device__ __forceinline__ float block_reduce_sum(float v, float* red) {
    float w = wave_reduce_sum(v);
    int lane = threadIdx.x & 31;
    int wid  = threadIdx.x >> 5;
    if (lane == 0) red[wid] = w;
    __syncthreads();
    float tot = 0.0f;
    int nw = blockDim.x >> 5;
    for (int i = 0; i < nw; ++i) tot += red[i];
    __syncthreads();
    return tot;
}

// Bilinear gather replicating the reference's clip-before-frac semantics.
__device__ __forceinline__ float interp2(const float* v, float xq, float yq) {
    int x0 = clampi((int)floorf(xq), 0, IMW - 1);
    int y0 = clampi((int)floorf(yq), 0, IMH - 1);
    int x1 = clampi((int)ceilf(xq), 0, IMW - 1);
    int y1 = clampi((int)ceilf(yq), 0, IMH - 1);
    float lh = yq - (float)y0;
    float lw = xq - (float)x0;
    float hh = 1.0f - lh, hw = 1.0f - lw;
    return v[y0 * IMW + x0] * (hh * hw) + v[y0 * IMW + x1] * (hh * lw) +
           v[y1 * IMW + x0] * (lh * hw) + v[y1 * IMW + x1] * (lh * lw);
}

// ---------------------------------------------------------------------------
// Image preprocessing
// ---------------------------------------------------------------------------
__global__ void k_gray(const float* __restrict__ prev,
                       const float* __restrict__ next,
                       float* __restrict__ img1f, float* __restrict__ img2f) {
    int i = blockIdx.x * blockDim.x + threadIdx.x;
    if (i >= HWPX) return;
    img1f[i] = 255.0f * (0.299f * prev[i] + 0.587f * prev[i + HWPX] +
                         0.114f * prev[i + 2 * HWPX]);
    img2f[i] = 255.0f * (0.299f * next[i] + 0.587f * next[i + HWPX] +
                         0.114f * next[i + 2 * HWPX]);
}

__global__ void k_blur_h(const float* __restrict__ src, float* __restrict__ dst) {
    int i = blockIdx.x * blockDim.x + threadIdx.x;
    if (i >= HWPX) return;
    int y = i / IMW, x = i - y * IMW;
    float acc = 0.0f;
#pragma unroll
    for (int d = -2; d <= 2; ++d) {
        int xi = x + d;
        xi = (xi < 0) ? -xi : ((xi >= IMW) ? (2 * IMW - 2 - xi) : xi);
        acc += KW5[d + 2] * src[y * IMW + xi];
    }
    dst[i] = acc;
}

__global__ void k_blur_v(const float* __restrict__ src, float* __restrict__ dst) {
    int i = blockIdx.x * blockDim.x + threadIdx.x;
    if (i >= HWPX) return;
    int y = i / IMW, x = i - y * IMW;
    float acc = 0.0f;
#pragma unroll
    for (int d = -2; d <= 2; ++d) {
        int yi = y + d;
        yi = (yi < 0) ? -yi : ((yi >= IMH) ? (2 * IMH - 2 - yi) : yi);
        acc += KW5[d + 2] * src[yi * IMW + x];
    }
    dst[i] = acc;  // dst = blur(gray1*255) == 255 * blur(gray1)
}

// jnp.gradient: central difference interior, one-sided at edges.
__global__ void k_grad(const float* __restrict__ Ib,
                       float* __restrict__ Ixm, float* __restrict__ Iym) {
    int i = blockIdx.x * blockDim.x + threadIdx.x;
    if (i >= HWPX) return;
    int y = i / IMW, x = i - y * IMW;
    {
        float xm = Ib[y * IMW + (x > 0 ? x - 1 : 0)];
        float xp = Ib[y * IMW + (x < IMW - 1 ? x + 1 : IMW - 1)];
        Ixm[i] = (x > 0 && x < IMW - 1) ? 0.5f * (xp - xm) : (xp - xm);
    }
    {
        float ym = Ib[(y > 0 ? y - 1 : 0) * IMW + x];
        float yp = Ib[(y < IMH - 1 ? y + 1 : IMH - 1) * IMW + x];
        Iym[i] = (y > 0 && y < IMH - 1) ? 0.5f * (yp - ym) : (yp - ym);
    }
}

// ---------------------------------------------------------------------------
// Track kernel: only points 0 and 1 determine the broadcast shifts
// (reference: X += sol[0,0]; Y += sol[1,0]).  One 256-thread block.
// ---------------------------------------------------------------------------
__global__ void __launch_bounds__(256)
k_track(const float* __restrict__ X0, const float* __restrict__ Y0,
        const float* __restrict__ img1f, const float* __restrict__ img2f,
        const float* __restrict__ Ixm, const float* __restrict__ Iym,
        float* __restrict__ shifts /* 2 floats */) {
    __shared__ float I1s[2][NSAMP], Jxs[2][NSAMP], Jys[2][NSAMP];
    __shared__ float red[8];
    __shared__ float AinvS[2][2];  // row 0 of Ainv only: [a00inv, a01inv]
    __shared__ float sX[2], sY[2], S[2];
    int tid = threadIdx.x;
    if (tid < 2) { sX[tid] = X0[tid]; sY[tid] = Y0[tid]; S[tid] = 0.0f; }
    __syncthreads();

    // Setup: window samples + Gram matrix + row 0 of 2x2 inverse, per point.
    for (int p = 0; p < 2; ++p) {
        float a00 = 0.0f, a01 = 0.0f, a11 = 0.0f;
        float px = sX[p], py = sY[p];
        for (int s = tid; s < NSAMP; s += blockDim.x) {
            float gx = (float)(s / WIN) + px - HALFW;  // meshgrid 'ij': mx = i
            float gy = (float)(s % WIN) + py - HALFW;
            float i1 = interp2(img1f, gx, gy);
            float jx = interp2(Ixm, gx, gy);
            float jy = interp2(Iym, gx, gy);
            I1s[p][s] = i1; Jxs[p][s] = jx; Jys[p][s] = jy;
            a00 += jx * jx; a01 += jx * jy; a11 += jy * jy;
        }
        a00 = block_reduce_sum(a00, red);
        a01 = block_reduce_sum(a01, red);
        a11 = block_reduce_sum(a11, red);
        if (tid == 0) {
            float invdet = 1.0f / (a00 * a11 - a01 * a01);
            AinvS[p][0] = a11 * invdet;   // Ainv[0][0]
            AinvS[p][1] = -a01 * invdet;  // Ainv[0][1]
        }
    }
    __syncthreads();

    // 15-level scan: all points share scalar increments (Sx from pt0, Sy from pt1).
    for (int t = 0; t < LEVELS; ++t) {
        float sol0[2];
        for (int p = 0; p < 2; ++p) {
            float b0 = 0.0f, b1 = 0.0f;
            float px = sX[p] + S[0], py = sY[p] + S[1];
            for (int s = tid; s < NSAMP; s += blockDim.x) {
                float gx = (float)(s / WIN) + px - HALFW;
                float gy = (float)(s % WIN) + py - HALFW;
                float d = interp2(img2f, gx, gy) - I1s[p][s];
                b0 -= Jxs[p][s] * d;
                b1 -= Jys[p][s] * d;
            }
            b0 = block_reduce_sum(b0, red);
            b1 = block_reduce_sum(b1, red);
            sol0[p] = AinvS[p][0] * b0 + AinvS[p][1] * b1;
        }
        if (tid == 0) { S[0] += sol0[0]; S[1] += sol0[1]; }
        __syncthreads();
    }
    if (tid == 0) { shifts[0] = S[0]; shifts[1] = S[1]; }
}

__global__ void k_apply(const float* __restrict__ X0, const float* __restrict__ Y0,
                        const float* __restrict__ shifts, float* __restrict__ out) {
    int i = blockIdx.x * blockDim.x + threadIdx.x;
    if (i >= NPTS) return;
    out[i]        = X0[i] + shifts[0];
    out[NPTS + i] = Y0[i] + shifts[1];
}

// ---------------------------------------------------------------------------
extern "C" void kernel_launch(void* const* d_in, const int* in_sizes, int n_in,
                              void* d_out, int out_size, void* d_ws, size_t ws_size,
                              hipStream_t stream) {
    const float* X0   = (const float*)d_in[0];  // (N)
    const float* Y0   = (const float*)d_in[1];  // (N)
    const float* prev = (const float*)d_in[2];  // (3,H,W)
    const float* next = (const float*)d_in[3];  // (3,H,W)
    float* out = (float*)d_out;                 // X (N) then Y (N)

    float* ws = (float*)d_ws;
    float* img1f = ws;             ws += HWPX;  // gray1 * 255
    float* img2f = ws;             ws += HWPX;  // gray2 * 255
    float* tmpH  = ws;             ws += HWPX;  // horizontal blur pass
    float* Ib    = ws;             ws += HWPX;  // 255 * blur(gray1)
    float* Ixm   = ws;             ws += HWPX;  // d/dx
    float* Iym   = ws;             ws += HWPX;  // d/dy
    float* shifts = ws;                         // 2 floats

    const int T = 256;
    const int Gimg = (HWPX + T - 1) / T;

    k_gray  <<<Gimg, T, 0, stream>>>(prev, next, img1f, img2f);
    k_blur_h<<<Gimg, T, 0, stream>>>(img1f, tmpH);
    k_blur_v<<<Gimg, T, 0, stream>>>(tmpH, Ib);
    k_grad  <<<Gimg, T, 0, stream>>>(Ib, Ixm, Iym);
    k_track <<<1, T, 0, stream>>>(X0, Y0, img1f, img2f, Ixm, Iym, shifts);
    k_apply <<<(NPTS + T - 1) / T, T, 0, stream>>>(X0, Y0, shifts, out);
}